// differN_noise_befor_21285857919120
// MI455X (gfx1250) — compile-verified
//
#include <hip/hip_runtime.h>
#include <hip/hip_bf16.h>
#include <cstdint>

typedef float v2f __attribute__((ext_vector_type(2)));
typedef float v8f __attribute__((ext_vector_type(8)));
typedef unsigned v4u __attribute__((ext_vector_type(4)));
typedef unsigned v8u __attribute__((ext_vector_type(8)));

#define NB    2048          // batch
#define PIX   784
#define WIRES 10
#define DIM   1024
#define KP    16            // padded component count (>=10)

// ---------------- workspace layout (in floats) ----------------
constexpr size_t OFF_MEAN  = 0;                         // 1024
constexpr size_t OFF_XC    = 1024;                      // 2048*784
constexpr size_t OFF_G     = OFF_XC   + 1605632;        // 784*784 (pad)
constexpr size_t OFF_V     = OFF_G    + 614912;         // 784*16
constexpr size_t OFF_W     = OFF_V    + 12544;          // 784*16 scratch
constexpr size_t OFF_Z     = OFF_W    + 12544;          // 2048*16
constexpr size_t OFF_GATES = OFF_Z    + 32768;          // 12*10*4*2 (pad)
constexpr size_t OFF_MRE   = OFF_GATES + 1024;          // 12*1024*1024
constexpr size_t OFF_MIM   = OFF_MRE  + 12582912;
constexpr size_t OFF_ARE   = OFF_MIM  + 12582912;       // 2048*1024
constexpr size_t OFF_AIM   = OFF_ARE  + 2097152;
constexpr size_t OFF_BRE   = OFF_AIM  + 2097152;
constexpr size_t OFF_BIM   = OFF_BRE  + 2097152;

// ---------------- PCA: column means + centering ----------------
__global__ void k_colmean(const float* __restrict__ x, float* __restrict__ mean) {
  int c = blockIdx.x * 256 + threadIdx.x;
  if (c >= PIX) return;
  float s = 0.f;
  for (int b = 0; b < NB; ++b) s += x[b * PIX + c];
  mean[c] = s * (1.0f / NB);
}

__global__ void k_center(const float* __restrict__ x, const float* __restrict__ mean,
                         float* __restrict__ xc) {
  int i = blockIdx.x * 256 + threadIdx.x;
  if (i >= NB * PIX) return;
  xc[i] = x[i] - mean[i % PIX];
}

// ---------------- Gram matrix G = xc^T * xc via WMMA f32 16x16x4 ----------------
__global__ __launch_bounds__(32) void k_gram(const float* __restrict__ xc,
                                             float* __restrict__ G) {
  const int lane = threadIdx.x;
  const int m0 = blockIdx.x * 16, n0 = blockIdx.y * 16;
  const int row = lane & 15;
  const int kk  = (lane >> 4) << 1;          // 0 or 2 (K half per lane group)
  v8f acc = {0.f,0.f,0.f,0.f,0.f,0.f,0.f,0.f};
  for (int k0 = 0; k0 < NB; k0 += 4) {
    v2f a, b;
    a.x = xc[(k0 + kk    ) * PIX + m0 + row];  // A[m][k] = xc[k][m]
    a.y = xc[(k0 + kk + 1) * PIX + m0 + row];
    b.x = xc[(k0 + kk    ) * PIX + n0 + row];  // B[k][n] = xc[k][n]
    b.y = xc[(k0 + kk + 1) * PIX + n0 + row];
    acc = __builtin_amdgcn_wmma_f32_16x16x4_f32(false, a, false, b, (short)0, acc,
                                                false, false);
  }
  const int cm = m0 + ((lane >> 4) << 3);
  const int cn = n0 + row;
#pragma unroll
  for (int v = 0; v < 8; ++v) G[(cm + v) * PIX + cn] = acc[v];
}

// ---------------- subspace (orthogonal) iteration for top-16 eigenvectors ----------------
__global__ __launch_bounds__(256) void k_subspace(const float* __restrict__ G,
                                                  float* __restrict__ Vout,
                                                  float* __restrict__ W) {
  __shared__ float lV[PIX * KP];
  __shared__ float red[256];
  const int tid = threadIdx.x;
  for (int idx = tid; idx < PIX * KP; idx += 256)
    lV[idx] = __sinf(0.5f + 0.37f * (float)idx);      // deterministic init
  __syncthreads();
  for (int it = 0; it < 16; ++it) {
    // W = G * V
    for (int idx = tid; idx < PIX * KP; idx += 256) {
      int r = idx >> 4, c = idx & 15;
      const float* gr = G + r * PIX;
      float acc = 0.f;
      for (int k = 0; k < PIX; ++k) acc += gr[k] * lV[k * KP + c];
      W[idx] = acc;
    }
    __syncthreads();
    // modified Gram-Schmidt: W -> lV
    for (int c = 0; c < KP; ++c) {
      for (int d = 0; d < c; ++d) {
        float p = 0.f;
        for (int r = tid; r < PIX; r += 256) p += lV[r * KP + d] * W[r * KP + c];
        red[tid] = p; __syncthreads();
        for (int o = 128; o > 0; o >>= 1) { if (tid < o) red[tid] += red[tid + o]; __syncthreads(); }
        float proj = red[0]; __syncthreads();
        for (int r = tid; r < PIX; r += 256) W[r * KP + c] -= proj * lV[r * KP + d];
        __syncthreads();
      }
      float p = 0.f;
      for (int r = tid; r < PIX; r += 256) { float v = W[r * KP + c]; p += v * v; }
      red[tid] = p; __syncthreads();
      for (int o = 128; o > 0; o >>= 1) { if (tid < o) red[tid] += red[tid + o]; __syncthreads(); }
      float inv = rsqrtf(fmaxf(red[0], 1e-20f)); __syncthreads();
      for (int r = tid; r < PIX; r += 256) lV[r * KP + c] = W[r * KP + c] * inv;
      __syncthreads();
    }
  }
  for (int idx = tid; idx < PIX * KP; idx += 256) Vout[idx] = lV[idx];
}

// ---------------- z = xc @ V  (2048x784x16) via WMMA ----------------
__global__ __launch_bounds__(32) void k_proj(const float* __restrict__ xc,
                                             const float* __restrict__ V,
                                             float* __restrict__ z) {
  const int lane = threadIdx.x;
  const int m0 = blockIdx.x * 16;
  const int row = lane & 15;
  const int kk  = (lane >> 4) << 1;
  v8f acc = {0.f,0.f,0.f,0.f,0.f,0.f,0.f,0.f};
  for (int k0 = 0; k0 < PIX; k0 += 4) {
    v2f a, b;
    a.x = xc[(m0 + row) * PIX + k0 + kk];
    a.y = xc[(m0 + row) * PIX + k0 + kk + 1];
    b.x = V[(k0 + kk    ) * KP + row];
    b.y = V[(k0 + kk + 1) * KP + row];
    acc = __builtin_amdgcn_wmma_f32_16x16x4_f32(false, a, false, b, (short)0, acc,
                                                false, false);
  }
  const int cm = m0 + ((lane >> 4) << 3);
  const int cn = row;
#pragma unroll
  for (int v = 0; v < 8; ++v) z[(cm + v) * KP + cn] = acc[v];
}

// ---------------- per-column sign fix (argmax |z| convention) ----------------
__global__ __launch_bounds__(256) void k_signfix(float* __restrict__ z) {
  const int c = blockIdx.x;        // 16 blocks
  const int tid = threadIdx.x;
  __shared__ float bv[256], bs[256];
  float best = -1.f, sv = 1.f;
  for (int b = tid; b < NB; b += 256) {
    float v = z[b * KP + c]; float a = fabsf(v);
    if (a > best) { best = a; sv = v; }
  }
  bv[tid] = best; bs[tid] = sv; __syncthreads();
  for (int o = 128; o > 0; o >>= 1) {
    if (tid < o && bv[tid + o] > bv[tid]) { bv[tid] = bv[tid + o]; bs[tid] = bs[tid + o]; }
    __syncthreads();
  }
  float s = (bs[0] < 0.f) ? -1.f : 1.f;
  __syncthreads();
  for (int b = tid; b < NB; b += 256) z[b * KP + c] *= s;
}

// ---------------- 2x2 gate matrices from weights ----------------
__global__ void k_gates(const float* __restrict__ w, float* __restrict__ gates) {
  int t = threadIdx.x;
  if (t >= 120) return;                   // 12 sublayer-matrices * 10 wires
  int m = t / 10, j = t % 10;
  const float* ww = w + (m * 10 + j) * 3; // weights (2,3,2,10,3) flat, m=((n*3+i)*2+l)
  float ph = ww[0], th = ww[1], om = ww[2];
  float c = cosf(0.5f * th), s = sinf(0.5f * th);
  float app = 0.5f * (ph + om), apm = 0.5f * (ph - om);
  float* g = gates + (size_t)(m * 10 + j) * 8;
  g[0] =  cosf(app) * c;  g[1] = -sinf(app) * c;   // U00 = e^{-i(phi+om)/2} c
  g[2] = -cosf(apm) * s;  g[3] = -sinf(apm) * s;   // U01 = -e^{+i(phi-om)/2} s
  g[4] =  cosf(apm) * s;  g[5] = -sinf(apm) * s;   // U10 = e^{-i(phi-om)/2} s
  g[6] =  cosf(app) * c;  g[7] =  sinf(app) * c;   // U11 = e^{+i(phi+om)/2} c
}

// ---------------- build dense sublayer unitaries Mt[s][s'] = cz[s'] * prod_j U_j ----------------
__global__ __launch_bounds__(256) void k_build(const float* __restrict__ gates,
                                               float* __restrict__ Mre,
                                               float* __restrict__ Mim) {
  const int m = blockIdx.y;                      // 0..11
  const int idx = blockIdx.x * 256 + threadIdx.x;
  const int sp = idx & (DIM - 1);                // output basis index s'
  const int s  = idx >> 10;                      // input basis index s
  float ar = 1.f, ai = 0.f;
#pragma unroll
  for (int j = 0; j < WIRES; ++j) {
    int bp = (sp >> (9 - j)) & 1;
    int bs = (s  >> (9 - j)) & 1;
    const float* g = gates + ((size_t)(m * 10 + j) * 4 + (bp * 2 + bs)) * 2;
    float gr = g[0], gi = g[1];
    float nr = ar * gr - ai * gi;
    ai = ar * gi + ai * gr;
    ar = nr;
  }
  // CZ ring phase on output index: ring offset r = (l==0)?1:2, l = m&1
  int r = (m & 1) + 1;
  int rot = ((sp << r) | (sp >> (WIRES - r))) & (DIM - 1);
  if (__popc(sp & rot) & 1) { ar = -ar; ai = -ai; }
  size_t o = (size_t)m * DIM * DIM + (size_t)s * DIM + sp;
  Mre[o] = ar; Mim[o] = ai;
}

// ---------------- state init |0> ----------------
__global__ void k_init_state(float* __restrict__ re, float* __restrict__ im) {
  int idx = blockIdx.x * 256 + threadIdx.x;     // 2048*1024 threads
  re[idx] = ((idx & (DIM - 1)) == 0) ? 1.f : 0.f;
  im[idx] = 0.f;
}

// ---------------- phase encoding: state *= exp(0.5i * z . sign(s)) ----------------
__global__ __launch_bounds__(256) void k_encode(const float* __restrict__ z,
                                                float* __restrict__ re,
                                                float* __restrict__ im) {
  __shared__ float lz[WIRES];
  const int b = blockIdx.x >> 2;
  const int s = ((blockIdx.x & 3) << 8) + threadIdx.x;
  if (threadIdx.x < WIRES) lz[threadIdx.x] = z[b * KP + threadIdx.x];
  __syncthreads();
  float ph = 0.f;
#pragma unroll
  for (int j = 0; j < WIRES; ++j) ph += ((s >> (9 - j)) & 1) ? lz[j] : -lz[j];
  ph *= 0.5f;
  float sn, cs;
  __sincosf(ph, &sn, &cs);
  int idx = b * DIM + s;
  float r0 = re[idx], i0 = im[idx];
  re[idx] = r0 * cs - i0 * sn;
  im[idx] = r0 * sn + i0 * cs;
}

// ---------------- main complex GEMM: C(2048x1024) = A(2048x1024) * B(1024x1024) ----------------
// 32(M) x 64(N) x 32(K) tiles. A tiles staged with per-lane async-to-LDS b128 copies
// (ASYNCcnt); B tiles staged with the Tensor Data Mover, one descriptor per plane
// issued by wave 0 (TENSORcnt). 8 waves, wave tile 16x16, WMMA f32 16x16x4.
__global__ __launch_bounds__(256) void k_cgemm(const float* __restrict__ Ar,
                                               const float* __restrict__ Ai,
                                               const float* __restrict__ Br,
                                               const float* __restrict__ Bi,
                                               float* __restrict__ Cr,
                                               float* __restrict__ Ci) {
  __shared__ float lAr[32][36], lAi[32][36];   // [m][k]; 36-f stride: 16B-aligned rows
  __shared__ float lBr[32][64], lBi[32][64];   // [k][n]; matches TDM packed tile
  const int tid  = threadIdx.x;
  const int lane = tid & 31;
  const int wave = tid >> 5;
  const int wm = wave & 1, wn = wave >> 1;     // 2(M) x 4(N) waves
  const int m0 = blockIdx.x * 32;
  const int n0 = blockIdx.y * 64;
  const int row = lane & 15;
  const int kk  = (lane >> 4) << 1;            // K pair select per lane half
  v8f acc_rr = {0.f,0.f,0.f,0.f,0.f,0.f,0.f,0.f};
  v8f acc_ii = {0.f,0.f,0.f,0.f,0.f,0.f,0.f,0.f};
  v8f acc_m  = {0.f,0.f,0.f,0.f,0.f,0.f,0.f,0.f};
  const int ar_row = tid >> 3, ar_k4 = (tid & 7) << 2;   // A staging map
  // constant part of the TDM D# group1: wg_mask=0, data_size=4B, tensor 1024x1024,
  // tile 64(dim0) x 32(dim1), dim0_stride=1024 elements
  const v8u g1 = { 0x20000u,                 // data_size=2 (4 bytes) at bits 17:16
                   (unsigned)(DIM << 16),    // tensor_dim0[15:0] at bits 63:48 (field 79:48)
                   (unsigned)(DIM << 16),    // tensor_dim1[15:0] at bits 95:80
                   (unsigned)(64 << 16),     // tile_dim0 = 64 at bits 127:112
                   32u,                      // tile_dim1 = 32 at bits 143:128
                   (unsigned)DIM,            // tensor_dim0_stride low32 = 1024
                   0u, 0u };
  const unsigned ldsBr = (unsigned)(uintptr_t)&lBr[0][0];
  const unsigned ldsBi = (unsigned)(uintptr_t)&lBi[0][0];
  for (int k0 = 0; k0 < DIM; k0 += 32) {
    // ---- B tiles via Tensor Data Mover (wave 0 issues; EXEC-independent DMA) ----
    if (wave == 0) {
      uint64_t gbr = (uint64_t)(uintptr_t)(Br + (size_t)k0 * DIM + n0);
      uint64_t gbi = (uint64_t)(uintptr_t)(Bi + (size_t)k0 * DIM + n0);
      v4u g0r = { 1u, ldsBr, (unsigned)gbr, (unsigned)(gbr >> 32) | (2u << 30) };
      v4u g0i = { 1u, ldsBi, (unsigned)gbi, (unsigned)(gbi >> 32) | (2u << 30) };
      asm volatile("tensor_load_to_lds %0, %1" :: "s"(g0r), "s"(g1) : "memory");
      asm volatile("tensor_load_to_lds %0, %1" :: "s"(g0i), "s"(g1) : "memory");
    }
    // ---- A tiles via per-lane async global->LDS b128 copies ----
    {
      unsigned lar = (unsigned)(uintptr_t)&lAr[ar_row][ar_k4];
      unsigned lai = (unsigned)(uintptr_t)&lAi[ar_row][ar_k4];
      uint64_t gar = (uint64_t)(uintptr_t)(Ar + (size_t)(m0 + ar_row) * DIM + k0 + ar_k4);
      uint64_t gai = (uint64_t)(uintptr_t)(Ai + (size_t)(m0 + ar_row) * DIM + k0 + ar_k4);
      asm volatile("global_load_async_to_lds_b128 %0, %1, off" :: "v"(lar), "v"(gar) : "memory");
      asm volatile("global_load_async_to_lds_b128 %0, %1, off" :: "v"(lai), "v"(gai) : "memory");
    }
    if (k0 + 32 < DIM)  // warm GL2 ahead of next TDM pull
      __builtin_prefetch(Br + (size_t)(k0 + 32) * DIM + n0, 0, 1);
    // ---- wait for our async copies (and TDM on wave 0), then block-sync ----
    asm volatile("s_wait_asynccnt 0x0" ::: "memory");
    if (wave == 0) asm volatile("s_wait_tensorcnt 0x0" ::: "memory");
    __syncthreads();
#pragma unroll
    for (int ks = 0; ks < 8; ++ks) {
      const int kl = ks * 4 + kk;
      v2f arf, aif, brf, bif;
      arf.x = lAr[wm * 16 + row][kl];     arf.y = lAr[wm * 16 + row][kl + 1];
      aif.x = lAi[wm * 16 + row][kl];     aif.y = lAi[wm * 16 + row][kl + 1];
      brf.x = lBr[kl][wn * 16 + row];     brf.y = lBr[kl + 1][wn * 16 + row];
      bif.x = lBi[kl][wn * 16 + row];     bif.y = lBi[kl + 1][wn * 16 + row];
      acc_rr = __builtin_amdgcn_wmma_f32_16x16x4_f32(false, arf, false, brf, (short)0, acc_rr, false, false);
      acc_ii = __builtin_amdgcn_wmma_f32_16x16x4_f32(false, aif, false, bif, (short)0, acc_ii, false, false);
      acc_m  = __builtin_amdgcn_wmma_f32_16x16x4_f32(false, arf, false, bif, (short)0, acc_m,  false, false);
      acc_m  = __builtin_amdgcn_wmma_f32_16x16x4_f32(false, aif, false, brf, (short)0, acc_m,  false, false);
    }
    __syncthreads();
  }
  const int cm = m0 + wm * 16 + ((lane >> 4) << 3);
  const int cn = n0 + wn * 16 + row;
#pragma unroll
  for (int v = 0; v < 8; ++v) {
    Cr[(size_t)(cm + v) * DIM + cn] = acc_rr[v] - acc_ii[v];  // Re = ArBr - AiBi
    Ci[(size_t)(cm + v) * DIM + cn] = acc_m[v];               // Im = ArBi + AiBr
  }
}

// ---------------- probabilities ----------------
__global__ void k_probs_z(const float* __restrict__ re, const float* __restrict__ im,
                          float* __restrict__ z) {
  int gid = blockIdx.x * 256 + threadIdx.x;    // 2048*16
  int b = gid >> 4, c = gid & 15;
  int idx = b * DIM + c;
  float r = re[idx], i = im[idx];
  z[gid] = r * r + i * i;
}

__global__ void k_probs_out(const float* __restrict__ re, const float* __restrict__ im,
                            float* __restrict__ out) {
  int gid = blockIdx.x * 256 + threadIdx.x;    // 2048*784
  int b = gid / PIX, p = gid % PIX;
  int idx = b * DIM + p;
  float r = re[idx], i = im[idx];
  out[gid] = fminf(fmaxf((r * r + i * i) * (float)PIX, 0.f), 1.f);
}

// ---------------- host launch ----------------
extern "C" void kernel_launch(void* const* d_in, const int* in_sizes, int n_in,
                              void* d_out, int out_size, void* d_ws, size_t ws_size,
                              hipStream_t stream) {
  (void)in_sizes; (void)n_in; (void)out_size; (void)ws_size;
  const float* x = (const float*)d_in[0];       // 2048*784
  const float* w = (const float*)d_in[1];       // 2*3*2*10*3 = 360
  float* out = (float*)d_out;
  float* ws  = (float*)d_ws;

  float* mean  = ws + OFF_MEAN;
  float* xc    = ws + OFF_XC;
  float* G     = ws + OFF_G;
  float* V     = ws + OFF_V;
  float* Wscr  = ws + OFF_W;
  float* z     = ws + OFF_Z;
  float* gates = ws + OFF_GATES;
  float* Mre   = ws + OFF_MRE;
  float* Mim   = ws + OFF_MIM;
  float* Are   = ws + OFF_ARE;
  float* Aim   = ws + OFF_AIM;
  float* Bre   = ws + OFF_BRE;
  float* Bim   = ws + OFF_BIM;

  // --- PCA ---
  k_colmean<<<4, 256, 0, stream>>>(x, mean);
  k_center<<<(NB * PIX) / 256, 256, 0, stream>>>(x, mean, xc);
  k_gram<<<dim3(PIX / 16, PIX / 16), 32, 0, stream>>>(xc, G);
  k_subspace<<<1, 256, 0, stream>>>(G, V, Wscr);
  k_proj<<<NB / 16, 32, 0, stream>>>(xc, V, z);
  k_signfix<<<KP, 256, 0, stream>>>(z);

  // --- build circuit sublayer unitaries ---
  k_gates<<<1, 128, 0, stream>>>(w, gates);
  k_build<<<dim3((DIM * DIM) / 256, 12), 256, 0, stream>>>(gates, Mre, Mim);

  // --- run circuit: 2 reps x 3 layers x 2 sublayer GEMMs ---
  dim3 gGrid(NB / 32, DIM / 64);
  for (int n = 0; n < 2; ++n) {
    k_init_state<<<(NB * DIM) / 256, 256, 0, stream>>>(Are, Aim);
    for (int i = 0; i < 3; ++i) {
      k_encode<<<NB * (DIM / 256), 256, 0, stream>>>(z, Are, Aim);
      size_t m = (size_t)((n * 3 + i) * 2) * DIM * DIM;
      k_cgemm<<<gGrid, 256, 0, stream>>>(Are, Aim, Mre + m, Mim + m, Bre, Bim);
      m += (size_t)DIM * DIM;
      k_cgemm<<<gGrid, 256, 0, stream>>>(Bre, Bim, Mre + m, Mim + m, Are, Aim);
    }
    if (n == 0)
      k_probs_z<<<(NB * KP) / 256, 256, 0, stream>>>(Are, Aim, z);
    else
      k_probs_out<<<(NB * PIX) / 256, 256, 0, stream>>>(Are, Aim, out);
  }
}